// NVPtriplane_79164837200470
// MI455X (gfx1250) — compile-verified
//
#include <hip/hip_runtime.h>
#include <math.h>

typedef __attribute__((ext_vector_type(2))) float v2f;
typedef __attribute__((ext_vector_type(8))) float v8f;

#define RESX 256
#define FDIM 32
#define HDIM 64
#define WAVES_PER_BLOCK 4
#define THREADS (WAVES_PER_BLOCK * 32)
#define FEAT_PAD 36   // 36*4=144 B row stride: 16B-aligned, conflict-free b128 stores
#define H_PAD 66      // 66 mod 64 = 2: conflict-free scalar row reads
#define W1_PAD 34     // 34*4=136 B: 8B-aligned b64 fragment reads, conflict-free

__device__ __forceinline__ float sigmoidf_(float z) {
    return 1.0f / (1.0f + expf(-z));
}

__launch_bounds__(THREADS)
__global__ void nvp_triplane_kernel(
    const float* __restrict__ t,       // (8,1)
    const float* __restrict__ x,       // (8,65536,3)
    const float* __restrict__ planes,  // (6,3,256,256,32)
    const float* __restrict__ w1g,     // (6,32,64)
    const float* __restrict__ b1g,     // (6,64)
    const float* __restrict__ w2g,     // (6,64,2)
    const float* __restrict__ b2g,     // (6,2)
    float* __restrict__ out)           // (8,65536,3)
{
    __shared__ __align__(16) float w1s[HDIM * W1_PAD];                       // [n][k] transposed
    __shared__ float b1s[HDIM];
    __shared__ float w2s[HDIM * 2];
    __shared__ float b2s[2];
    __shared__ __align__(16) float featS[WAVES_PER_BLOCK][32 * FEAT_PAD];    // [pt][k]
    __shared__ __align__(16) float hS[WAVES_PER_BLOCK][32 * H_PAD];          // [pt][n]

    const int tid  = threadIdx.x;
    const int lane = tid & 31;
    const int wave = tid >> 5;
    const int lmod = lane & 15;
    const int lhi  = lane >> 4;

    const int p    = blockIdx.x * THREADS + tid;   // one point per thread; grid exactly covers B*N
    const int bidx = p >> 16;                      // N = 65536

    float y0 = x[p * 3 + 0];
    float y1 = x[p * 3 + 1];
    float y2 = x[p * 3 + 2];
    const float tn = t[bidx] * 2.0f - 1.0f;

    for (int l = 0; l < 6; ++l) {
        __syncthreads();  // previous layer's LDS consumers done before overwrite

        // ---- stage layer weights in LDS (w1 transposed to [n][k]) ----
        {
            const float* w1l = w1g + l * FDIM * HDIM;
            for (int e = tid; e < FDIM * HDIM; e += THREADS) {
                int k = e >> 6;       // w1 layout (K=32, N=64) row-major
                int n = e & 63;
                w1s[n * W1_PAD + k] = w1l[e];
            }
            if (tid < HDIM)     b1s[tid] = b1g[l * HDIM + tid];
            if (tid < HDIM * 2) w2s[tid] = w2g[l * HDIM * 2 + tid];
            if (tid < 2)        b2s[tid] = b2g[l * 2 + tid];
        }

        // ---- coupling inputs: sel = l % 3, keep = other two dims ----
        const int sel = (l >= 3) ? (l - 3) : l;
        float a0, a1;
        if (sel == 0)      { a0 = y1; a1 = y2; }
        else if (sel == 1) { a0 = y0; a1 = y2; }
        else               { a0 = y0; a1 = y1; }
        const float u = tanhf(a0);
        const float v = tanhf(a1);

        // ---- gather: 3 bilinear plane samples summed into feat[32] ----
        float4 acc[8];
        #pragma unroll
        for (int j = 0; j < 8; ++j) acc[j] = make_float4(0.f, 0.f, 0.f, 0.f);

        const float* plbase = planes + (size_t)(l * 3) * (RESX * RESX * FDIM);
        const float us[3] = { u, u, v };
        const float vs[3] = { v, tn, tn };

        #pragma unroll
        for (int q = 0; q < 3; ++q) {
            float gx = fminf(fmaxf((us[q] + 1.0f) * 0.5f * (float)(RESX - 1), 0.0f), (float)(RESX - 1));
            float gy = fminf(fmaxf((vs[q] + 1.0f) * 0.5f * (float)(RESX - 1), 0.0f), (float)(RESX - 1));
            float fx = floorf(gx), fy = floorf(gy);
            int xi0 = (int)fx, yi0 = (int)fy;
            int xi1 = min(xi0 + 1, RESX - 1);
            int yi1 = min(yi0 + 1, RESX - 1);
            float wx = gx - fx, wy = gy - fy;
            float w00 = (1.0f - wx) * (1.0f - wy);
            float w01 = wx * (1.0f - wy);
            float w10 = (1.0f - wx) * wy;
            float w11 = wx * wy;

            const float* pl = plbase + q * (RESX * RESX * FDIM);
            const float4* c00 = (const float4*)(pl + ((size_t)yi0 * RESX + xi0) * FDIM);
            const float4* c01 = (const float4*)(pl + ((size_t)yi0 * RESX + xi1) * FDIM);
            const float4* c10 = (const float4*)(pl + ((size_t)yi1 * RESX + xi0) * FDIM);
            const float4* c11 = (const float4*)(pl + ((size_t)yi1 * RESX + xi1) * FDIM);

            #pragma unroll
            for (int j = 0; j < 8; ++j) {
                float4 f00 = c00[j], f01 = c01[j], f10 = c10[j], f11 = c11[j];
                acc[j].x += w00 * f00.x + w01 * f01.x + w10 * f10.x + w11 * f11.x;
                acc[j].y += w00 * f00.y + w01 * f01.y + w10 * f10.y + w11 * f11.y;
                acc[j].z += w00 * f00.z + w01 * f01.z + w10 * f10.z + w11 * f11.z;
                acc[j].w += w00 * f00.w + w01 * f01.w + w10 * f10.w + w11 * f11.w;
            }
        }

        // ---- stage feat tile (32 pts x 32 feats) in LDS ----
        {
            float4* fr = (float4*)&featS[wave][lane * FEAT_PAD];
            #pragma unroll
            for (int j = 0; j < 8; ++j) fr[j] = acc[j];
        }
        __syncthreads();  // w1s + featS visible

        // ---- WMMA: h(32pts x 64) = feat(32pts x 32) @ w1(32 x 64), fp32 ----
        {
            const float* fw = &featS[wave][0];
            float*       hw = &hS[wave][0];
            #pragma unroll
            for (int mt = 0; mt < 2; ++mt) {
                #pragma unroll
                for (int nt = 0; nt < 4; ++nt) {
                    v8f c = {};
                    #pragma unroll
                    for (int s = 0; s < 8; ++s) {
                        const int k0 = s * 4 + lhi * 2;  // lanes 16-31 hold K+2,K+3
                        v2f a = *(const v2f*)&fw[(mt * 16 + lmod) * FEAT_PAD + k0];
                        v2f b = *(const v2f*)&w1s[(nt * 16 + lmod) * W1_PAD + k0];
                        c = __builtin_amdgcn_wmma_f32_16x16x4_f32(
                                false, a, false, b, (short)0, c, false, false);
                    }
                    // D layout: VGPR j -> row j (lanes 0-15) / row j+8 (lanes 16-31), col = lane%16
                    #pragma unroll
                    for (int j = 0; j < 8; ++j)
                        hw[(mt * 16 + j + lhi * 8) * H_PAD + nt * 16 + lmod] = c[j];
                }
            }
        }
        __syncthreads();

        // ---- per-point epilogue: bias, leaky relu, h @ w2, sigmoid, update ----
        {
            const float* hr = &hS[wave][lane * H_PAD];
            float z0 = b2s[0], z1 = b2s[1];
            #pragma unroll
            for (int j = 0; j < HDIM; ++j) {
                float hv = hr[j] + b1s[j];
                hv = (hv > 0.0f) ? hv : 0.01f * hv;
                z0 += hv * w2s[j * 2 + 0];
                z1 += hv * w2s[j * 2 + 1];
            }
            const float st0 = sigmoidf_(z0);
            const float st1 = sigmoidf_(z1);
            const float s  = 5.0f * st0 + 0.005f;
            const float tt = st1 * 6.0f - 3.0f;
            if (sel == 0)      y0 = (y0 - tt) / s;
            else if (sel == 1) y1 = (y1 - tt) / s;
            else               y2 = (y2 - tt) / s;
        }
    }

    out[p * 3 + 0] = y0;
    out[p * 3 + 1] = y1;
    out[p * 3 + 2] = y2;
}

extern "C" void kernel_launch(void* const* d_in, const int* in_sizes, int n_in,
                              void* d_out, int out_size, void* d_ws, size_t ws_size,
                              hipStream_t stream) {
    const float* t      = (const float*)d_in[0];
    // d_in[1] = f : unused by the reference computation
    const float* x      = (const float*)d_in[2];
    const float* planes = (const float*)d_in[3];
    const float* w1     = (const float*)d_in[4];
    const float* b1     = (const float*)d_in[5];
    const float* w2     = (const float*)d_in[6];
    const float* b2     = (const float*)d_in[7];
    float* out = (float*)d_out;

    const int total_points = in_sizes[2] / 3;           // B*N = 524288
    const int blocks = total_points / THREADS;          // 4096 (exact)
    nvp_triplane_kernel<<<blocks, THREADS, 0, stream>>>(
        t, x, planes, w1, b1, w2, b2, out);
}